// SupervisedContrastiveLoss_49692771614953
// MI455X (gfx1250) — compile-verified
//
#include <hip/hip_runtime.h>
#include <hip/hip_bf16.h>

// Supervised contrastive loss, fully fused for MI455X (gfx1250, wave32).
//
// Each 256-thread block (8 wave32s) owns a 16-row strip of sim = E*E^T.
// It stages its 16 embedding rows (16x256 f32 = 16KB) in LDS and makes 5
// passes over all 8192 columns, recomputing the strip each pass with
// V_WMMA_F32_16X16X4_F32 (f32 is mandatory: exp(20*sim) amplifies bf16
// error by ~e^2). Embeddings are 8MB -> fully L2-resident (192MB L2), so
// 5x recompute costs ~20GB of L2 reads, ~zero HBM, and only 2KB workspace
// vs 256MB for a materialized sim matrix.
//
// Per wave, two adjacent 16-col tiles are computed simultaneously with two
// independent accumulator chains: each K-chunk issues 1 ds_load_b64 (A,
// shared) + 2 global_load_b64 (B) + 2 v_wmma, letting the XDL pipeline
// overlap the two D->C dependency chains and halving LDS traffic per FLOP.
//
// Pass 0: pos_sum accumulation + byte-0 histogram of sortable(sim_masked).
// Pass 1..3: accumulate exp() for values strictly above the known byte
//            prefix; histogram the next byte for prefix-equal values.
// Pass 4: accumulate exp() for the last byte above the threshold.
// Finalize: neg_sum += r * exp(t/T) for the r ties at the exact k-th value
// (bit-exact top-k semantics). loss_i = -log(pos/(pos+neg)); block partials
// -> d_ws, single fixed-order reduction -> d_out (deterministic per replay).

#define N_EMB   8192
#define D_EMB   256
#define K_SEL   1638            // max(3, int(8192*0.2))
#define INV_T   20.0f           // 1/0.05
#define NEG_FILL (-1e9f)
#define TILES   (N_EMB / 16)    // 512 column tiles
#define PAIRS   (TILES / 2)     // 256 tile pairs (2 tiles per wave iter)
#define ROWBLKS (N_EMB / 16)    // 512 row blocks

typedef __attribute__((ext_vector_type(2))) float v2f;
typedef __attribute__((ext_vector_type(8))) float v8f;

__device__ __forceinline__ unsigned toSortable(float f) {
  unsigned s = __float_as_uint(f);
  return (s & 0x80000000u) ? ~s : (s | 0x80000000u);
}
__device__ __forceinline__ float fromSortable(unsigned u) {
  unsigned s = (u & 0x80000000u) ? (u & 0x7fffffffu) : ~u;
  return __uint_as_float(s);
}

__global__ __launch_bounds__(256) void scl_main(const float* __restrict__ emb,
                                                const int* __restrict__ labels,
                                                float* __restrict__ partial) {
  __shared__ float    sA[16 * D_EMB];     // 16KB: this block's 16 rows of E
  __shared__ unsigned sHist[16 * 256];    // 16KB: per-row radix histograms
  __shared__ int      sRowLab[16];
  __shared__ float    sPos[16];           // sum exp(sim/T) over positives
  __shared__ float    sNeg[16];           // sum exp over confirmed-top negatives
  __shared__ unsigned sNpos[16];          // count of positives (same label, j!=i)
  __shared__ unsigned sPref[16];          // sortable-u32 threshold prefix
  __shared__ int      sKrem[16];          // remaining k within current prefix bin
  __shared__ float    sLoss[16];

  const int tid     = threadIdx.x;
  const int lane    = tid & 31;
  const int wave    = tid >> 5;
  const int half    = lane >> 4;   // 0: lanes 0-15, 1: lanes 16-31
  const int nIdx    = lane & 15;
  const int rowbase = blockIdx.x * 16;

  // Stage A strip (contiguous 16*256 floats) into LDS.
  {
    const float4* src = (const float4*)(emb + (size_t)rowbase * D_EMB);
    float4* dst = (float4*)sA;
    for (int i = tid; i < (16 * D_EMB) / 4; i += 256) dst[i] = src[i];
  }
  if (tid < 16) {
    sRowLab[tid] = labels[rowbase + tid];
    sPos[tid] = 0.f; sNeg[tid] = 0.f; sNpos[tid] = 0u;
    sPref[tid] = 0u; sKrem[tid] = K_SEL;
  }
  for (int i = tid; i < 16 * 256; i += 256) sHist[i] = 0u;
  __syncthreads();

  for (int p = 0; p < 5; ++p) {
    // Each wave handles tile pairs pt = wave, wave+8, ... (32 pairs/wave).
    for (int pt = wave; pt < PAIRS; pt += 8) {
      const int colbase0 = pt * 32;        // first 16-col tile
      const int colbase1 = colbase0 + 16;  // second 16-col tile
      // A (16x4) and B (4x16) operands share the same f32 WMMA lane layout
      // because sim = E*E^T is symmetric: float2 at row (base + lane&15),
      // column offset kk + 2*(lane>>4).
      const float* ap  = sA  + (size_t)nIdx * D_EMB + half * 2;
      const float* bp0 = emb + (size_t)(colbase0 + nIdx) * D_EMB + half * 2;
      const float* bp1 = emb + (size_t)(colbase1 + nIdx) * D_EMB + half * 2;
      if (pt + 8 < PAIRS)  // uniform per wave: prefetch next pair's B rows
        __builtin_prefetch(emb + (size_t)(colbase0 + 256 + nIdx) * D_EMB, 0, 1);

      v8f acc0 = {0.f, 0.f, 0.f, 0.f, 0.f, 0.f, 0.f, 0.f};
      v8f acc1 = {0.f, 0.f, 0.f, 0.f, 0.f, 0.f, 0.f, 0.f};
      #pragma unroll 8
      for (int kk = 0; kk < D_EMB; kk += 4) {
        v2f a  = *(const v2f*)(ap  + kk);  // ds_load_b64 (shared by 2 WMMAs)
        v2f b0 = *(const v2f*)(bp0 + kk);  // global_load_b64 (L2-resident)
        v2f b1 = *(const v2f*)(bp1 + kk);
        acc0 = __builtin_amdgcn_wmma_f32_16x16x4_f32(
            false, a, false, b0, (short)0, acc0, false, false);
        acc1 = __builtin_amdgcn_wmma_f32_16x16x4_f32(
            false, a, false, b1, (short)0, acc1, false, false);
      }

      // C layout: VGPR r holds (M = r + 8*half, N = lane&15). The column
      // (and thus its label) is constant across all 8 accumulator regs.
      #pragma unroll
      for (int t = 0; t < 2; ++t) {
        const v8f& acc = (t == 0) ? acc0 : acc1;
        const int  col  = (t == 0 ? colbase0 : colbase1) + nIdx;
        const int  lcol = labels[col];
        #pragma unroll
        for (int r = 0; r < 8; ++r) {
          const int   m  = r + 8 * half;
          const float v  = acc[r];
          const bool  eq = (lcol == sRowLab[m]);
          if (p == 0 && eq && (col != rowbase + m)) {
            atomicAdd(&sPos[m], __expf(v * INV_T));   // ds_add_f32
            atomicAdd(&sNpos[m], 1u);
          }
          const float    vn = eq ? NEG_FILL : v;      // reference's masked_fill
          const unsigned u  = toSortable(vn);
          if (p == 0) {
            atomicAdd(&sHist[m * 256 + (u >> 24)], 1u);
          } else {
            const unsigned pref     = sPref[m];
            const unsigned maskPrev = (p == 1) ? 0u
                                               : (0xFFFFFFFFu << (40 - 8 * p));
            const int      sc       = 32 - 8 * p;     // shift of byte (p-1)
            if ((u & maskPrev) == (pref & maskPrev)) {
              const unsigned ub = (u >> sc) & 0xFFu;
              const unsigned pb = (pref >> sc) & 0xFFu;
              if (ub > pb) {
                atomicAdd(&sNeg[m], __expf(vn * INV_T));
              } else if (ub == pb && p <= 3) {
                atomicAdd(&sHist[m * 256 + ((u >> (sc - 8)) & 0xFFu)], 1u);
              }
            }
          }
        }
      }
    }
    __syncthreads();
    // Radix-select scan: thread t owns row t; find the byte bin containing
    // the k-th largest, fold strictly-above counts into krem, reset hist.
    if (p < 4 && tid < 16) {
      const int need = sKrem[tid];
      unsigned cum = 0;
      int b;
      for (b = 255; b >= 0; --b) {
        const unsigned c = sHist[tid * 256 + b];
        if (cum + c >= (unsigned)need) break;
        cum += c;
      }
      if (b < 0) b = 0;  // unreachable: every row has 8192 candidates >= k
      sPref[tid] |= (unsigned)b << (24 - 8 * p);
      sKrem[tid]  = need - (int)cum;
      for (int i = 0; i < 256; ++i) sHist[tid * 256 + i] = 0u;
    }
    __syncthreads();
  }

  if (tid < 16) {
    const float thr  = fromSortable(sPref[tid]);
    const float negs = sNeg[tid] + (float)sKrem[tid] * __expf(thr * INV_T);
    // masked pos entries each contribute exp(0)=1 (faithful to reference)
    const float poss = sPos[tid] + (float)(N_EMB - (int)sNpos[tid]);
    sLoss[tid] = -__logf(poss / (poss + negs));
  }
  __syncthreads();
  if (tid == 0) {
    float s = 0.f;
    for (int i = 0; i < 16; ++i) s += sLoss[i];
    partial[blockIdx.x] = s * (1.0f / (float)N_EMB);
  }
}

__global__ void scl_reduce(const float* __restrict__ partial,
                           float* __restrict__ out) {
  if (blockIdx.x == 0 && threadIdx.x == 0) {
    float s = 0.f;
    for (int i = 0; i < ROWBLKS; ++i) s += partial[i];  // fixed order: deterministic
    out[0] = s;
  }
}

extern "C" void kernel_launch(void* const* d_in, const int* in_sizes, int n_in,
                              void* d_out, int out_size, void* d_ws, size_t ws_size,
                              hipStream_t stream) {
  const float* emb    = (const float*)d_in[0];
  const int*   labels = (const int*)d_in[1];
  float*       out    = (float*)d_out;
  float*       wsf    = (float*)d_ws;   // needs ROWBLKS * 4 = 2KB

  scl_main<<<ROWBLKS, 256, 0, stream>>>(emb, labels, wsf);
  scl_reduce<<<1, 32, 0, stream>>>(wsf, out);
}